// K_algorithm_43336220016940
// MI455X (gfx1250) — compile-verified
//
#include <hip/hip_runtime.h>
#include <cstdint>

#define B_      128
#define L_      8192
#define F_      9
#define THREADS 1024
#define PER     (L_ / THREADS)   // 8
#define NWAVES  (THREADS / 32)   // 32

__device__ __forceinline__ double wave_incl_scan(double v, int lane) {
    #pragma unroll
    for (int d = 1; d < 32; d <<= 1) {
        double u = __shfl_up(v, d, 32);
        if (lane >= d) v += u;
    }
    return v;
}

__global__ __launch_bounds__(THREADS, 1)
void kenneth_kernel(const float* __restrict__ conc,
                    const float* __restrict__ kern,
                    float* __restrict__ out)
{
    // ~225 KB of the WGP's 320 KB LDS: the whole per-batch working set.
    __shared__ float  bcd_s[L_];      // 32 KB
    __shared__ double S_s[L_];        // 64 KB  prefix sum of row sums
    __shared__ double G_s[L_];        // 64 KB  prefix sum of g
    __shared__ double H_s[L_];        // 64 KB  prefix sum of h
    __shared__ double wsumA[NWAVES];  // 256 B
    __shared__ double wsumB[NWAVES];  // 256 B

    const int b    = blockIdx.x;
    const int tid  = threadIdx.x;
    const int lane = tid & 31;
    const int wv   = tid >> 5;
    const float* base = conc + (size_t)b * L_ * F_;

    // ---------- Phase 1: row sums + thread-local inclusive scan.
    // bcd column is DMA'd global->LDS with CDNA5 async loads (ASYNCcnt),
    // overlapped with the VALU row-sum stream.
    double locS[PER];
    double run = 0.0;
    {
        const unsigned long long gbase = (unsigned long long)base;
        #pragma unroll
        for (int i = 0; i < PER; ++i) {
            const int r = tid * PER + i;
            unsigned lds_addr = (unsigned)(uintptr_t)(const void*)&bcd_s[r];
            unsigned goff     = (unsigned)(r * F_ * 4);
            asm volatile("global_load_async_to_lds_b32 %0, %1, %2"
                         :: "v"(lds_addr), "v"(goff), "s"(gbase) : "memory");
            const float* row = base + (size_t)r * F_;
            double rs = 0.0;
            #pragma unroll
            for (int f = 0; f < F_; ++f) rs += (double)row[f];
            run += rs;
            locS[i] = run;
        }
    }

    // ---------- Phase 2: hierarchical block scan (wave shuffles, 3 barriers)
    {
        double v = wave_incl_scan(run, lane);
        if (lane == 31) wsumA[wv] = v;
        double lexcl = __shfl_up(v, 1, 32);
        if (lane == 0) lexcl = 0.0;

        asm volatile("s_wait_asynccnt 0x0" ::: "memory");  // bcd_s resident
        __syncthreads();
        if (tid < 32) {                       // wave 0 scans the 32 wave totals
            double w = wave_incl_scan(wsumA[lane], lane);
            wsumA[lane] = w;
        }
        __syncthreads();
        double off = ((wv > 0) ? wsumA[wv - 1] : 0.0) + lexcl;
        #pragma unroll
        for (int i = 0; i < PER; ++i)
            S_s[tid * PER + i] = off + locS[i];
    }
    __syncthreads();

    const double Stot = S_s[L_ - 1];

    // ---------- Phase 3: g(m)=bcd*W_f, h(m)=bcd*W_r and their scans.
    double locG[PER], locH[PER];
    {
        double rg = 0.0, rh = 0.0;
        #pragma unroll
        for (int i = 0; i < PER; ++i) {
            const int r = tid * PER + i;
            const double bv = (double)bcd_s[r];
            const double wf = (r >= 14)      ? (1.0 + S_s[r - 14])        : 1.0;
            const double wr = (r <= L_ - 15) ? (1.0 + Stot - S_s[r + 13]) : 1.0;
            rg += bv * wf; locG[i] = rg;
            rh += bv * wr; locH[i] = rh;
        }
        double vg = wave_incl_scan(rg, lane);
        double vh = wave_incl_scan(rh, lane);
        if (lane == 31) { wsumA[wv] = vg; wsumB[wv] = vh; }
        double leg = __shfl_up(vg, 1, 32);
        double leh = __shfl_up(vh, 1, 32);
        if (lane == 0) { leg = 0.0; leh = 0.0; }
        __syncthreads();
        if (tid < 32) {
            double wa = wave_incl_scan(wsumA[lane], lane);
            double wb = wave_incl_scan(wsumB[lane], lane);
            wsumA[lane] = wa; wsumB[lane] = wb;
        }
        __syncthreads();
        double eg = ((wv > 0) ? wsumA[wv - 1] : 0.0) + leg;
        double eh = ((wv > 0) ? wsumB[wv - 1] : 0.0) + leh;
        #pragma unroll
        for (int i = 0; i < PER; ++i) {
            G_s[tid * PER + i] = eg + locG[i];
            H_s[tid * PER + i] = eh + locH[i];
        }
    }
    __syncthreads();

    // ---------- Phase 4: elementwise outputs, coalesced over t.
    const double invZ1 = 1.0 / (1.0 + Stot);
    const double kk    = (double)kern[0];
    float* ob = out + (size_t)b * F_ * L_;
    #pragma unroll
    for (int i = 0; i < PER; ++i) {
        const int r = i * THREADS + tid;
        const float bv = bcd_s[r];
        double gsum = ((r >= 16) ? G_s[r - 16] : 0.0)
                    - ((r >= 76) ? G_s[r - 76] : 0.0);
        int hi = r + 75; if (hi > L_ - 1) hi = L_ - 1;
        int lo = r + 15; if (lo > L_ - 1) lo = L_ - 1;
        double hsum = H_s[hi] - H_s[lo];
        double csum = kk * (double)bv * (gsum + hsum);      // c_fwd + c_rev
        double num  = (double)bv * (csum + (double)bv);
        float  a0   = (bv + 1.0f) - ((bv > 1e-13f) ? 1.0f : 0.0f); // f32, like ref
        float  f0   = (float)(num / (double)a0 * invZ1);
        f0 = fminf(fmaxf(f0, 0.0f), 0.9999f);
        ob[r] = f0;

        const float* row = base + (size_t)r * F_;
        #pragma unroll
        for (int j = 1; j < F_; ++j) {
            float v  = row[j];
            float aj = (v + 1.0f) - ((v > 1e-13f) ? 1.0f : 0.0f);
            float fj = (float)((double)v * (double)v / (double)aj * invZ1);
            fj = fminf(fmaxf(fj, 0.0f), 0.9999f);
            ob[(size_t)j * L_ + r] = fj;
        }
    }
}

extern "C" void kernel_launch(void* const* d_in, const int* in_sizes, int n_in,
                              void* d_out, int out_size, void* d_ws, size_t ws_size,
                              hipStream_t stream) {
    (void)in_sizes; (void)n_in; (void)d_ws; (void)ws_size; (void)out_size;
    const float* conc = (const float*)d_in[0];
    const float* kern = (const float*)d_in[1];
    float* out = (float*)d_out;
    kenneth_kernel<<<B_, THREADS, 0, stream>>>(conc, kern, out);
}